// AE_29789893165628
// MI455X (gfx1250) — compile-verified
//
#include <hip/hip_runtime.h>
#include <hip/hip_bf16.h>

typedef __attribute__((ext_vector_type(2))) float v2f;
typedef __attribute__((ext_vector_type(8))) float v8f;

#define SSTRIDE 16384
#define NTILES_SIDE 515
#define NBLOCKS_GRP (2 * NTILES_SIDE)
#define SCALE_QK 0.44721359549995793f
#define EPS_LN 1e-5f

__device__ __forceinline__ v8f wmma4(v2f a, v2f b, v8f c) {
  // V_WMMA_F32_16X16X4_F32 : D(16x16) = A(16x4) * B(4x16) + C
  return __builtin_amdgcn_wmma_f32_16x16x4_f32(false, a, false, b, (short)0, c,
                                               false, false);
}
__device__ __forceinline__ v2f mk2(float a, float b) { v2f r; r.x = a; r.y = b; return r; }

__device__ __forceinline__ void matvec5(const float* __restrict__ w,
                                        const float* __restrict__ b,
                                        const float* x, float* y) {
#pragma unroll
  for (int o = 0; o < 5; ++o) {
    float s = b[o];
#pragma unroll
    for (int i = 0; i < 5; ++i) s = fmaf(w[o * 5 + i], x[i], s);
    y[o] = s;
  }
}

__device__ __forceinline__ void layernorm5(const float* x,
                                           const float* __restrict__ g,
                                           const float* __restrict__ b, float* y) {
  float m = 0.f;
#pragma unroll
  for (int i = 0; i < 5; ++i) m += x[i];
  m *= 0.2f;
  float v = 0.f;
#pragma unroll
  for (int i = 0; i < 5; ++i) { float d = x[i] - m; v = fmaf(d, d, v); }
  v *= 0.2f;
  float r = rsqrtf(v + EPS_LN);
#pragma unroll
  for (int i = 0; i < 5; ++i) y[i] = fmaf((x[i] - m) * r, g[i], b[i]);
}

// group g = (level, side): level L in [1,13], n = 2^(L-1) rows,
// node(row) = (2^L - 1 + side) + 2*row   (side 0 = left/odd, 1 = right/even)
__device__ __forceinline__ void decode_group(int b, int& lvl, int& side, int& tile) {
  side = 0;
  if (b >= NTILES_SIDE) { side = 1; b -= NTILES_SIDE; }
  int acc = 0;
  tile = 0;
#pragma unroll 1
  for (lvl = 1; lvl <= 13; ++lvl) {
    int nt = (lvl <= 5) ? 1 : (1 << (lvl - 5));
    if (b < acc + nt) { tile = b - acc; return; }
    acc += nt;
  }
  lvl = 13;
}

// ---------------------------------------------------------------- prep: q/k/v
__global__ void prep_kernel(
    const float* __restrict__ X,
    const float* wqe, const float* bqe, const float* wke, const float* bke,
    const float* wve, const float* bve,
    const float* wqd, const float* bqd, const float* wkd, const float* bkd,
    const float* wvd, const float* bvd,
    float* qe, float* ke, float* ve, float* qd, float* kd, float* vd,
    float* accum) {
  int i = blockIdx.x * blockDim.x + threadIdx.x;
  if (i < 32) accum[i] = 0.f;
  if (i < 1 || i > 16382) return;
  float x[5], t[5];
#pragma unroll
  for (int d = 0; d < 5; ++d) x[d] = X[i * 5 + d];
  matvec5(wqe, bqe, x, t);
#pragma unroll
  for (int d = 0; d < 5; ++d) qe[i * 5 + d] = t[d];
  matvec5(wke, bke, x, t);
#pragma unroll
  for (int d = 0; d < 5; ++d) ke[i * 5 + d] = t[d];
  matvec5(wve, bve, x, t);
#pragma unroll
  for (int d = 0; d < 5; ++d) ve[i * 5 + d] = t[d];
  matvec5(wqd, bqd, x, t);
#pragma unroll
  for (int d = 0; d < 5; ++d) qd[i * 5 + d] = t[d];
  matvec5(wkd, bkd, x, t);
#pragma unroll
  for (int d = 0; d < 5; ++d) kd[i * 5 + d] = t[d];
  matvec5(wvd, bvd, x, t);
#pragma unroll
  for (int d = 0; d < 5; ++d) vd[i * 5 + d] = t[d];
}

// ------------------------- pack FFN weights into WMMA B-fragment layout
// w1p: 4 segments x 2048 float2:
//   seg0[col] = {w1[col,0], w1[col,1]}   (lanes 0-15, K-chunk 0)
//   seg1[col] = {w1[col,2], w1[col,3]}   (lanes 16-31, K-chunk 0)
//   seg2[col] = {w1[col,4], 0}           (lanes 0-15, K-chunk 1)
//   seg3[col] = {0, 0}                   (lanes 16-31, K-chunk 1)
// w2p: [kpair 0..1023][n 0..15] float2 = {w2[n, 2p], w2[n, 2p+1]}, n>=5 -> 0
__global__ void pack_kernel(const float* __restrict__ w1,
                            const float* __restrict__ w2,
                            v2f* __restrict__ w1p, v2f* __restrict__ w2p) {
  int i = blockIdx.x * blockDim.x + threadIdx.x;
  if (i < 2048) {
    const float* wc = w1 + (size_t)i * 5;
    w1p[i] = mk2(wc[0], wc[1]);
    w1p[2048 + i] = mk2(wc[2], wc[3]);
    w1p[2 * 2048 + i] = mk2(wc[4], 0.f);
    w1p[3 * 2048 + i] = mk2(0.f, 0.f);
  }
  if (i < 1024 * 16) {
    const int p = i >> 4, n = i & 15;
    v2f v = mk2(0.f, 0.f);
    if (n < 5) v = mk2(w2[(size_t)n * 2048 + 2 * p], w2[(size_t)n * 2048 + 2 * p + 1]);
    w2p[i] = v;
  }
}

// -------------------------------------------- attention + residual + LN (+ opt matvec)
// one wave (32 threads) per 16-row tile; lane = (row 0..15, column-partition 0..1)
__global__ __launch_bounds__(32) void attn_kernel(
    const float* __restrict__ qarr, const float* __restrict__ karr,
    const float* __restrict__ varr, const float* __restrict__ wo,
    const float* __restrict__ bo, const float* __restrict__ resid,
    const float* __restrict__ lng, const float* __restrict__ lnb,
    float* __restrict__ outarr, const float* __restrict__ xw,
    const float* __restrict__ xb, float* __restrict__ xout) {
  int lvl, side, tile;
  decode_group((int)blockIdx.x, lvl, side, tile);
  const int n = 1 << (lvl - 1);
  const int base = (1 << lvl) - 1 + side;
  const int row0 = tile * 16;

  const int lane = (int)threadIdx.x;
  const int r = lane & 15;
  const int part = lane >> 4;
  const int grow = row0 + r;
  const bool active = grow < n;
  const int node = base + 2 * grow;

  float q[5] = {0.f, 0.f, 0.f, 0.f, 0.f};
  if (active) {
#pragma unroll
    for (int d = 0; d < 5; ++d) q[d] = qarr[node * 5 + d];
  }

  float mmax = -3.0e38f, ssum = 0.f;
  float acc[5] = {0.f, 0.f, 0.f, 0.f, 0.f};
  for (int j = part; j < n; j += 2) {
    const int cn = base + 2 * j;
    const float* kj = karr + (size_t)cn * 5;
    float s = 0.f;
#pragma unroll
    for (int d = 0; d < 5; ++d) s = fmaf(q[d], kj[d], s);
    s *= SCALE_QK;
    const float mn = fmaxf(mmax, s);
    const float corr = __expf(mmax - mn);
    const float p = __expf(s - mn);
    ssum = ssum * corr + p;
    const float* vj = varr + (size_t)cn * 5;
#pragma unroll
    for (int d = 0; d < 5; ++d) acc[d] = fmaf(p, vj[d], acc[d] * corr);
    mmax = mn;
  }
  // merge the two column partitions (lane <-> lane^16)
  const float m2 = __shfl_xor(mmax, 16, 32);
  const float s2 = __shfl_xor(ssum, 16, 32);
  const float mn = fmaxf(mmax, m2);
  const float c1 = __expf(mmax - mn);
  const float c2 = __expf(m2 - mn);
  const float ws = ssum * c1 + s2 * c2;
  float av[5];
#pragma unroll
  for (int d = 0; d < 5; ++d) {
    const float a2 = __shfl_xor(acc[d], 16, 32);
    av[d] = (acc[d] * c1 + a2 * c2) / ws;
  }

  float o[5];
  matvec5(wo, bo, av, o);
  float e0[5] = {0.f, 0.f, 0.f, 0.f, 0.f};
  if (active) {
#pragma unroll
    for (int d = 0; d < 5; ++d) e0[d] = resid[node * 5 + d] + o[d];
  }
  float y[5];
  layernorm5(e0, lng, lnb, y);
  if (part == 0 && active) {
#pragma unroll
    for (int d = 0; d < 5; ++d) outarr[node * 5 + d] = y[d];
    if (xout) {
      float t[5];
      matvec5(xw, xb, y, t);
#pragma unroll
      for (int d = 0; d < 5; ++d) xout[node * 5 + d] = t[d];
    }
  }
}

// -------------------------------- FFN (WMMA) + residual + 2x LN (+ mem/kv or loss)
__global__ __launch_bounds__(32) void ffn_kernel(
    const float* __restrict__ inarr, const v2f* __restrict__ w1p,
    const float* __restrict__ b1, const v2f* __restrict__ w2p,
    const float* __restrict__ b2, const float* __restrict__ gA,
    const float* __restrict__ bA, const float* __restrict__ gB,
    const float* __restrict__ bB, const float* __restrict__ X,
    float* __restrict__ outarr, const float* __restrict__ wk,
    const float* __restrict__ bk, const float* __restrict__ wv,
    const float* __restrict__ bv, float* __restrict__ kout,
    float* __restrict__ vout, float* __restrict__ accum, int mode) {
  __shared__ float stage[16 * 17];

  int lvl, side, tile;
  decode_group((int)blockIdx.x, lvl, side, tile);
  const int n = 1 << (lvl - 1);
  const int base = (1 << lvl) - 1 + side;
  const int row0 = tile * 16;

  const int lane = (int)threadIdx.x;
  const int m = lane & 15;   // A-layout row (and B/C column)
  const int hi = lane >> 4;  // K half select
  const int grow = row0 + m;
  const bool active = grow < n;
  const int node = base + 2 * grow;

  float xin[5] = {0.f, 0.f, 0.f, 0.f, 0.f};
  if (active) {
#pragma unroll
    for (int d = 0; d < 5; ++d) xin[d] = inarr[node * 5 + d];
  }
  // A (16x5 padded to 16x8): lanes 0-15 K={0,1}, lanes 16-31 K={2,3}; chunk1 K={4,pad}
  const v2f a0 = mk2(hi ? xin[2] : xin[0], hi ? xin[3] : xin[1]);
  const v2f a1 = mk2(hi ? 0.f : xin[4], 0.f);

  const v2f* w1c0 = w1p + (size_t)hi * 2048;        // seg 0/1
  const v2f* w1c1 = w1p + (size_t)(2 + hi) * 2048;  // seg 2/3 (seg3 = zeros)

  v8f facc = {};
  for (int nb = 0; nb < 128; ++nb) {
    const int col = nb * 16 + m;  // hidden neuron index
    const v2f b0 = w1c0[col];
    const v2f b1v = w1c1[col];
    v8f h = {};
    h = wmma4(a0, b0, h);   // K = 0..3
    h = wmma4(a1, b1v, h);  // K = 4 (padded)
    const float bias = b1[col];
#pragma unroll
    for (int p = 0; p < 8; ++p) {
      const float t = h[p] + bias;
      stage[(p + 8 * hi) * 17 + m] = t > 0.f ? t : 0.f;  // relu, C row = p+8*hi
    }
    __syncthreads();
    // GEMM2 partial: f(16x5) += h_tile(16x16) @ w2_chunk(16x16-padded)
#pragma unroll
    for (int kc = 0; kc < 4; ++kc) {
      const int k0 = 4 * kc + 2 * hi;
      const v2f a2 = mk2(stage[m * 17 + k0], stage[m * 17 + k0 + 1]);
      const v2f bm = w2p[(size_t)(nb * 8 + 2 * kc + hi) * 16 + m];
      facc = wmma4(a2, bm, facc);
    }
    __syncthreads();
  }
  // spill f tile to LDS so each row lives in one lane
#pragma unroll
  for (int p = 0; p < 8; ++p) stage[(p + 8 * hi) * 17 + m] = facc[p];
  __syncthreads();

  if (hi == 0 && active) {
    float f[5], t[5], e2[5], y[5];
#pragma unroll
    for (int d = 0; d < 5; ++d) f[d] = stage[m * 17 + d] + b2[d];
#pragma unroll
    for (int d = 0; d < 5; ++d) t[d] = xin[d] + f[d];
    layernorm5(t, gA, bA, e2);
    layernorm5(e2, gB, bB, y);
    if (mode == 0) {
#pragma unroll
      for (int d = 0; d < 5; ++d) outarr[node * 5 + d] = y[d];
      float kc5[5], vc5[5];
      matvec5(wk, bk, y, kc5);
      matvec5(wv, bv, y, vc5);
#pragma unroll
      for (int d = 0; d < 5; ++d) {
        kout[node * 5 + d] = kc5[d];
        vout[node * 5 + d] = vc5[d];
      }
    } else {
      float diff = 0.f;
#pragma unroll
      for (int d = 0; d < 5; ++d) diff += fabsf(X[node * 5 + d] - y[d]);
      atomicAdd(&accum[(lvl - 1) + 13 * side], diff);
    }
  }
}

// ---------------------------------------------------------------- finalize
__global__ __launch_bounds__(32) void finalize_kernel(const float* __restrict__ accum,
                                                      float* __restrict__ out) {
  __shared__ float means[26];
  const int t = (int)threadIdx.x;
  if (t < 26) {
    const int lvl = (t % 13) + 1;
    const int n = 1 << (lvl - 1);
    const float mean = accum[t] / (float)(n * 5);
    out[1 + t] = mean;  // out[1..13]=lcheck, out[14..26]=rcheck
    means[t] = mean;
  }
  __syncthreads();
  if (t == 0) {
    float s = 0.f;
    for (int j = 0; j < 26; ++j) s += means[j];
    out[0] = s;
  }
}

extern "C" void kernel_launch(void* const* d_in, const int* in_sizes, int n_in,
                              void* d_out, int out_size, void* d_ws, size_t ws_size,
                              hipStream_t stream) {
  (void)in_sizes; (void)n_in; (void)out_size; (void)ws_size;
  // JAX pytree (sorted-dict) flattening of setup_inputs():
  enum {
    DEC_B1 = 1, DEC_B2, DEC_CA_BK, DEC_CA_BO, DEC_CA_BQ, DEC_CA_BV,
    DEC_CA_WK, DEC_CA_WO, DEC_CA_WQ, DEC_CA_WV,
    DEC_N1_G, DEC_N1_B, DEC_N2_G, DEC_N2_B, DEC_N3_G, DEC_N3_B,
    DEC_SA_BK, DEC_SA_BO, DEC_SA_BQ, DEC_SA_BV,
    DEC_SA_WK, DEC_SA_WO, DEC_SA_WQ, DEC_SA_WV,
    DEC_W1, DEC_W2, DEC_NORM_G, DEC_NORM_B,
    ENC_B1, ENC_B2, ENC_N1_G, ENC_N1_B, ENC_N2_G, ENC_N2_B,
    ENC_SA_BK, ENC_SA_BO, ENC_SA_BQ, ENC_SA_BV,
    ENC_SA_WK, ENC_SA_WO, ENC_SA_WQ, ENC_SA_WV,
    ENC_W1, ENC_W2, ENC_NORM_G, ENC_NORM_B
  };
  const float* X = (const float*)d_in[0];
  auto F = [&](int i) { return (const float*)d_in[i]; };

  float* ws = (float*)d_ws;
  const size_t A = (size_t)SSTRIDE * 5;
  float* qe = ws + 0 * A;  float* ke = ws + 1 * A;  float* ve = ws + 2 * A;
  float* qd = ws + 3 * A;  float* kd = ws + 4 * A;  float* vd = ws + 5 * A;
  float* e1 = ws + 6 * A;  float* mem = ws + 7 * A;
  float* kca = ws + 8 * A; float* vca = ws + 9 * A; float* qca = ws + 10 * A;
  float* d1 = ws + 11 * A; float* d2 = ws + 12 * A;
  float* accum = ws + 13 * A;  // 32 floats
  // packed weight area (8-byte aligned: offsets are even float counts)
  v2f* w1p_enc = (v2f*)(ws + 13 * A + 32);
  v2f* w2p_enc = w1p_enc + 4 * 2048;
  v2f* w1p_dec = w2p_enc + 1024 * 16;
  v2f* w2p_dec = w1p_dec + 4 * 2048;

  prep_kernel<<<64, 256, 0, stream>>>(
      X, F(ENC_SA_WQ), F(ENC_SA_BQ), F(ENC_SA_WK), F(ENC_SA_BK), F(ENC_SA_WV),
      F(ENC_SA_BV), F(DEC_SA_WQ), F(DEC_SA_BQ), F(DEC_SA_WK), F(DEC_SA_BK),
      F(DEC_SA_WV), F(DEC_SA_BV), qe, ke, ve, qd, kd, vd, accum);
  pack_kernel<<<64, 256, 0, stream>>>(F(ENC_W1), F(ENC_W2), w1p_enc, w2p_enc);
  pack_kernel<<<64, 256, 0, stream>>>(F(DEC_W1), F(DEC_W2), w1p_dec, w2p_dec);

  // encoder self-attn -> e1 = LN(x + sa(x), enc.n1)
  attn_kernel<<<NBLOCKS_GRP, 32, 0, stream>>>(
      qe, ke, ve, F(ENC_SA_WO), F(ENC_SA_BO), X, F(ENC_N1_G), F(ENC_N1_B), e1,
      nullptr, nullptr, nullptr);
  // encoder FFN -> mem = LN(LN(e1+ffn(e1), n2), enc_norm); also k_ca/v_ca = mem@W
  ffn_kernel<<<NBLOCKS_GRP, 32, 0, stream>>>(
      e1, w1p_enc, F(ENC_B1), w2p_enc, F(ENC_B2), F(ENC_N2_G), F(ENC_N2_B),
      F(ENC_NORM_G), F(ENC_NORM_B), nullptr, mem, F(DEC_CA_WK), F(DEC_CA_BK),
      F(DEC_CA_WV), F(DEC_CA_BV), kca, vca, nullptr, 0);
  // decoder self-attn -> d1 = LN(x + sa(x), dec.n1); also q_ca = d1@Wq_ca
  attn_kernel<<<NBLOCKS_GRP, 32, 0, stream>>>(
      qd, kd, vd, F(DEC_SA_WO), F(DEC_SA_BO), X, F(DEC_N1_G), F(DEC_N1_B), d1,
      F(DEC_CA_WQ), F(DEC_CA_BQ), qca);
  // decoder cross-attn -> d2 = LN(d1 + ca(d1, mem), dec.n2)
  attn_kernel<<<NBLOCKS_GRP, 32, 0, stream>>>(
      qca, kca, vca, F(DEC_CA_WO), F(DEC_CA_BO), d1, F(DEC_N2_G), F(DEC_N2_B), d2,
      nullptr, nullptr, nullptr);
  // decoder FFN -> out = LN(LN(d2+ffn(d2), n3), dec_norm); loss accumulation
  ffn_kernel<<<NBLOCKS_GRP, 32, 0, stream>>>(
      d2, w1p_dec, F(DEC_B1), w2p_dec, F(DEC_B2), F(DEC_N3_G), F(DEC_N3_B),
      F(DEC_NORM_G), F(DEC_NORM_B), X, nullptr, nullptr, nullptr, nullptr,
      nullptr, nullptr, nullptr, accum, 1);

  finalize_kernel<<<1, 32, 0, stream>>>(accum, (float*)d_out);
}